// HashEncoding_721554506107
// MI455X (gfx1250) — compile-verified
//
#include <hip/hip_runtime.h>

// CDNA5 / gfx1250, wave32.
typedef float v2f __attribute__((ext_vector_type(2)));
typedef float v4f __attribute__((ext_vector_type(4)));
typedef float v8f __attribute__((ext_vector_type(8)));

#define HE_NLEVELS    16
#define HE_STARTHASH  6
#define HE_TPRIME     1048583u
#define HE_DENSETOTAL 822944

// ---------------------------------------------------------------------------
// Pass 1: row sums of a (nrows x 16) f32 matrix via V_WMMA_F32_16X16X4_F32.
// A = all-ones (16x4), B = 4x16 chunk of 16 table rows (rows -> matrix cols).
// D[m,n] = sum_k B[k,n]; accumulating 4 WMMAs over K covers all 16 features,
// so D[0,n] = rowsum(row0+n). Because A is all-ones, the result is invariant
// to the K-slot permutation inside the B operand layout (sum over K).
// Each wave: lane r (r<16) loads cols 0..7 of row (row0+r) as two float4s,
// lane r+16 loads cols 8..15 -> 1KB fully-coalesced per 16-row group.
//
// Table reads are NON-TEMPORAL (global_load_b128 th:NT): the 724 MB stream is
// never reused, and we must not evict the 45 MB of row-sum output that pass 2
// re-reads from the 192 MB L2. Row-sum stores stay regular-temporal.
// Memory-bound pass (~31 us at 23.3 TB/s); the 4 WMMAs per KB are free.
// ---------------------------------------------------------------------------
__global__ __launch_bounds__(256)
void he_rowsum16_wmma(const float* __restrict__ src, float* __restrict__ dst,
                      int nrows) {
    const int lane = threadIdx.x & 31;
    const int half = lane >> 4;        // 0 -> cols 0..7, 1 -> cols 8..15
    const int r15  = lane & 15;
    int wave   = (int)((blockIdx.x * blockDim.x + threadIdx.x) >> 5);
    int nwaves = (int)((gridDim.x * blockDim.x) >> 5);
    int ngroups = (nrows + 15) >> 4;

    const v2f a_ones = {1.0f, 1.0f};

    for (int g = wave; g < ngroups; g += nwaves) {
        int row0 = g << 4;
        int row  = row0 + r15;

        v4f d0 = {0.f, 0.f, 0.f, 0.f};
        v4f d1 = {0.f, 0.f, 0.f, 0.f};
        if (row < nrows) {
            const v4f* p = reinterpret_cast<const v4f*>(
                src + (size_t)row * 16 + (size_t)half * 8);
            d0 = __builtin_nontemporal_load(p);      // b128, th:NT
            d1 = __builtin_nontemporal_load(p + 1);  // b128, th:NT
        }

        // WMMAs at top level: EXEC is all-ones here (full 256-thread blocks).
        v8f acc = {};
        v2f b;
        b.x = d0.x; b.y = d0.y;
        acc = __builtin_amdgcn_wmma_f32_16x16x4_f32(false, a_ones, false, b,
                                                    (short)0, acc, false, false);
        b.x = d0.z; b.y = d0.w;
        acc = __builtin_amdgcn_wmma_f32_16x16x4_f32(false, a_ones, false, b,
                                                    (short)0, acc, false, false);
        b.x = d1.x; b.y = d1.y;
        acc = __builtin_amdgcn_wmma_f32_16x16x4_f32(false, a_ones, false, b,
                                                    (short)0, acc, false, false);
        b.x = d1.z; b.y = d1.w;
        acc = __builtin_amdgcn_wmma_f32_16x16x4_f32(false, a_ones, false, b,
                                                    (short)0, acc, false, false);

        // D VGPR0: lanes 0..15 hold D[0, n=lane] = rowsum(row0+lane).
        if (lane < 16 && row < nrows) dst[row0 + lane] = acc[0];
    }
}

// ---------------------------------------------------------------------------
// Pass 2: one thread per (point, level). 8 gathers of 4B row sums from the
// 45 MB table that now lives in L2 (192 MB). Index math replicates the
// reference bit-for-bit (float division by es, float add of corner offsets
// before truncation, clamps, int64 hash).
// ---------------------------------------------------------------------------
__global__ __launch_bounds__(256)
void he_gather(const float* __restrict__ xyz, const float* __restrict__ rowsum,
               float* __restrict__ out, int npts) {
    // Per-level constants (computed from BASE_RES=16, B=1.38).
    const int NTAB[HE_NLEVELS] = {16, 22, 30, 42, 58, 80, 110, 152,
                                  210, 290, 400, 553, 763, 1053, 1453, 2005};
    // es = float32( 1.0 / (n-1) ) computed in double like numpy.
    const float ESTAB[HE_NLEVELS] = {
        (float)(1.0 / 15.0),   (float)(1.0 / 21.0),   (float)(1.0 / 29.0),
        (float)(1.0 / 41.0),   (float)(1.0 / 57.0),   (float)(1.0 / 79.0),
        (float)(1.0 / 109.0),  (float)(1.0 / 151.0),  (float)(1.0 / 209.0),
        (float)(1.0 / 289.0),  (float)(1.0 / 399.0),  (float)(1.0 / 552.0),
        (float)(1.0 / 762.0),  (float)(1.0 / 1052.0), (float)(1.0 / 1452.0),
        (float)(1.0 / 2004.0)};
    const int DOFF[HE_STARTHASH] = {0, 4096, 14744, 41744, 115832, 310944};

    int tid = (int)(blockIdx.x * blockDim.x + threadIdx.x);
    if (tid >= npts * HE_NLEVELS) return;
    int p = tid >> 4;
    int l = tid & 15;

    float x = xyz[p * 3 + 0];
    float y = xyz[p * 3 + 1];
    float z = xyz[p * 3 + 2];
    // (x - lo) / (hi - lo) with lo=-2, hi=2; /4 == *0.25 exactly.
    float xnx = (x + 2.0f) * 0.25f;
    float xny = (y + 2.0f) * 0.25f;
    float xnz = (z + 2.0f) * 0.25f;

    const int   n   = NTAB[l];
    const float es  = ESTAB[l];
    const int   nm1 = n - 1;

    float fx = xnx / es;  // true IEEE division, matches reference
    float fy = xny / es;
    float fz = xnz / es;

    // base corner = clip(trunc(f), 0, n-1); f >= 0 always.
    int bx0 = min((int)fx, nm1);
    int by0 = min((int)fy, nm1);
    int bz0 = min((int)fz, nm1);
    // +1 corner = clip(trunc(f + 1.0f), 0, n-1)  (float add THEN truncate).
    int bx1 = min((int)(fx + 1.0f), nm1);
    int by1 = min((int)(fy + 1.0f), nm1);
    int bz1 = min((int)(fz + 1.0f), nm1);

    float ox = fx - (float)bx0;
    float oy = fy - (float)by0;

    // Reference weight: w = mul_x * mul_y (z weight intentionally absent).
    float wx[2] = {1.0f - ox, ox};
    float wy[2] = {1.0f - oy, oy};
    int ix[2] = {bx0, bx1};
    int iy[2] = {by0, by1};
    int iz[2] = {bz0, bz1};

    float acc = 0.0f;
    if (l < HE_STARTHASH) {
        const float* tab = rowsum + DOFF[l];
#pragma unroll
        for (int c = 0; c < 8; ++c) {  // OFFSETS order: bit2=x, bit1=y, bit0=z
            int cx = ix[(c >> 2) & 1];
            int cy = iy[(c >> 1) & 1];
            int cz = iz[c & 1];
            int idx = (cx * n + cy) * n + cz;
            acc += wx[(c >> 2) & 1] * wy[(c >> 1) & 1] * tab[idx];
        }
    } else {
        const float* tab =
            rowsum + HE_DENSETOTAL + (size_t)(l - HE_STARTHASH) * HE_TPRIME;
#pragma unroll
        for (int c = 0; c < 8; ++c) {
            unsigned long long cx = (unsigned long long)ix[(c >> 2) & 1];
            unsigned long long cy = (unsigned long long)iy[(c >> 1) & 1];
            unsigned long long cz = (unsigned long long)iz[c & 1];
            unsigned long long h =
                cx ^ (cy * 19349663ull) ^ (cz * 83492791ull);
            unsigned int hi = (unsigned int)(h % (unsigned long long)HE_TPRIME);
            acc += wx[(c >> 2) & 1] * wy[(c >> 1) & 1] * tab[hi];
        }
    }

    float* po = out + (size_t)p * 19;
    po[3 + l] = acc;
    if (l == 0) {
        po[0] = xnx;
        po[1] = xny;
        po[2] = xnz;
    }
}

extern "C" void kernel_launch(void* const* d_in, const int* in_sizes, int n_in,
                              void* d_out, int out_size, void* d_ws,
                              size_t ws_size, hipStream_t stream) {
    const float* xyz   = (const float*)d_in[0];  // (N, 3) f32
    const float* dense = (const float*)d_in[1];  // (822944, 16) f32
    const float* hasht = (const float*)d_in[2];  // (10, 1048583, 16) f32
    float* out = (float*)d_out;
    float* ws  = (float*)d_ws;  // row sums: 822944 + 10*1048583 floats (~45 MB)

    int npts       = in_sizes[0] / 3;
    int dense_rows = in_sizes[1] / 16;   // 822944
    int hash_rows  = in_sizes[2] / 16;   // 10485830

    // Pass 1: row sums via WMMA. 8 waves/block, 1 wave per 16-row group.
    {
        int groups = (dense_rows + 15) >> 4;
        int blocks = (groups + 7) >> 3;
        he_rowsum16_wmma<<<blocks, 256, 0, stream>>>(dense, ws, dense_rows);
    }
    {
        int groups = (hash_rows + 15) >> 4;
        int blocks = (groups + 7) >> 3;
        he_rowsum16_wmma<<<blocks, 256, 0, stream>>>(hasht, ws + dense_rows,
                                                     hash_rows);
    }

    // Pass 2: gather + bilinear(x,y) weights from L2-resident row sums.
    {
        int total  = npts * HE_NLEVELS;
        int blocks = (total + 255) / 256;
        he_gather<<<blocks, 256, 0, stream>>>(xyz, ws, out, npts);
    }
}